// PathEncoder_76897094467927
// MI455X (gfx1250) — compile-verified
//
#include <hip/hip_runtime.h>

// PathEncoder for MI455X (gfx1250, wave32).
//
// out[b,x,y,h] = sum_{l,d} edata[b, path[b,x,y,l], d] * emb[l,h,d] / clip(dist,1,L)
//
// Memory-bound gather kernel (~256MB of L2-resident traffic vs ~1 GFLOP).
// Exact-precision f32 WMMA (V_WMMA_F32_16X16X4_F32), K=256 in 64 steps of 4.
// One wave = one 16(pos) x 16(head, 8 used) tile.
//
// Round-2 changes vs round-1 (which compiled but serialized):
//  * no EXEC branching in the hot loop: pad rows handled by clamped index +
//    value select; B fragments come from a pre-zeroed per-lane LDS table.
//  * per path element l: batch-load 16 A fragments (16 global_load_b64 in
//    flight per lane) + 16 B fragments, then 16 back-to-back WMMAs.
//  * two accumulators (even/odd step) to break the WMMA accumulate chain.

typedef __attribute__((ext_vector_type(2))) float v2f;
typedef __attribute__((ext_vector_type(8))) float v8f;

#define MAX_LEN   4
#define NUM_HEADS 8
#define FEAT_DIM  64
#define B_SZ      16
#define N_SZ      128
#define E_SZ      4096

#define NSTEP     (MAX_LEN * FEAT_DIM / 4)   // 64 K-steps of 4
#define STEPS_L   (FEAT_DIM / 4)             // 16 steps per path element

__global__ __launch_bounds__(256) void path_encoder_wmma(
    const float* __restrict__ edge_feat,   // [B, E, 64] f32
    const int*   __restrict__ path,        // [B, N, N, 4] i32, values in [0, E]
    const int*   __restrict__ dist,        // [B, N, N] i32
    const float* __restrict__ emb_weight,  // [32, 64] f32
    float*       __restrict__ out)         // [B, N, N, 8] f32
{
    // Per-lane B-fragment table: s_bf[step][lane] = {B[k0+2h][col], B[k0+2h+1][col]},
    // zero for col >= 8. One unconditional ds_load_b64 per step, conflict-free.
    __shared__ v2f s_bf[NSTEP * 32];                  // 16 KB

    for (int e = threadIdx.x; e < NSTEP * 32; e += blockDim.x) {
        const int s  = e >> 5;                        // step 0..63
        const int ln = e & 31;
        const int c  = ln & 15;
        const int h  = ln >> 4;
        const int l  = s >> 4;                        // path element
        const int dd = (s & 15) << 2;                 // K base within row
        v2f v; v.x = 0.0f; v.y = 0.0f;
        if (c < NUM_HEADS) {
            const float* p = emb_weight + (l * NUM_HEADS + c) * FEAT_DIM + dd + 2 * h;
            v.x = p[0];
            v.y = p[1];
        }
        s_bf[e] = v;
    }
    __syncthreads();

    const int lane = threadIdx.x & 31;
    const int wave = threadIdx.x >> 5;
    const int col  = lane & 15;   // A row (tile-local position), B/D column (head)
    const int half = lane >> 4;   // selects K-pair (A/B) and M-half (D)

    // 8 waves/block, 16 positions/wave -> 128 positions per block.
    const long long p0  = (long long)blockIdx.x * 128 + (long long)wave * 16;
    const long long pos = p0 + col;                       // flat (b,x,y)
    const int b = (int)(pos / (long long)(N_SZ * N_SZ));

    // 4 path indices for this lane's position (lanes 16..31 duplicate 0..15).
    const int4 pidx = *(const int4*)(path + pos * MAX_LEN);
    const int idxs[MAX_LEN] = { pidx.x, pidx.y, pidx.z, pidx.w };

    const float* ebase = edge_feat + (size_t)b * E_SZ * FEAT_DIM + 2 * half;

    v8f acc0 = {};
    v8f acc1 = {};

#pragma unroll
    for (int l = 0; l < MAX_LEN; ++l) {
        const int  row   = idxs[l];
        const bool valid = (row < E_SZ);              // row == E -> zero pad row
        const int  crow  = valid ? row : 0;           // safe row: load unconditionally
        const float* src = ebase + (size_t)crow * FEAT_DIM;

        // Batch A gathers: 16 independent global_load_b64 per lane.
        v2f a[STEPS_L];
#pragma unroll
        for (int s = 0; s < STEPS_L; ++s)
            a[s] = *(const v2f*)(src + 4 * s);

        // Batch B fragments from the LDS table.
        v2f bf[STEPS_L];
#pragma unroll
        for (int s = 0; s < STEPS_L; ++s)
            bf[s] = s_bf[(l * STEPS_L + s) * 32 + lane];

        // Zero out pad-row contributions (v_cndmask, no EXEC divergence).
#pragma unroll
        for (int s = 0; s < STEPS_L; ++s) {
            a[s].x = valid ? a[s].x : 0.0f;
            a[s].y = valid ? a[s].y : 0.0f;
        }

        // 16 WMMAs, alternating accumulators to break the RAW chain.
#pragma unroll
        for (int s = 0; s < STEPS_L; s += 2) {
            acc0 = __builtin_amdgcn_wmma_f32_16x16x4_f32(
                false, a[s],     false, bf[s],     (short)0, acc0, false, false);
            acc1 = __builtin_amdgcn_wmma_f32_16x16x4_f32(
                false, a[s + 1], false, bf[s + 1], (short)0, acc1, false, false);
        }
    }

    const v8f acc = acc0 + acc1;

    // D layout: acc[r] = D[M = r + 8*half][N = col]; only heads col < 8 are real.
    if (col < NUM_HEADS) {
#pragma unroll
        for (int r = 0; r < 8; ++r) {
            const long long opos = p0 + r + 8 * half;
            int dv = dist[opos];
            dv = dv < 1 ? 1 : (dv > MAX_LEN ? MAX_LEN : dv);
            out[opos * NUM_HEADS + col] = acc[r] / (float)dv;
        }
    }
}

extern "C" void kernel_launch(void* const* d_in, const int* in_sizes, int n_in,
                              void* d_out, int out_size, void* d_ws, size_t ws_size,
                              hipStream_t stream) {
    const float* edge_feat  = (const float*)d_in[0];  // [16,4096,64] f32
    const int*   path       = (const int*)  d_in[1];  // [16,128,128,4] i32
    const int*   dist       = (const int*)  d_in[2];  // [16,128,128] i32
    const float* emb_weight = (const float*)d_in[3];  // [32,64] f32
    float*       out        = (float*)d_out;          // [16,128,128,8] f32

    // 262144 positions total, 128 per block (8 waves x 16).
    const int total_pos = B_SZ * N_SZ * N_SZ;
    const int blocks    = total_pos / 128;            // 2048
    path_encoder_wmma<<<blocks, 256, 0, stream>>>(edge_feat, path, dist,
                                                  emb_weight, out);
}